// ESALSTM_1030792151669
// MI455X (gfx1250) — compile-verified
//
#include <hip/hip_runtime.h>
#include <math.h>

typedef __attribute__((ext_vector_type(2))) float v2f;
typedef __attribute__((ext_vector_type(8))) float v8f;

#define E 5
#define H 30

__device__ __forceinline__ float sigf(float x) { return 1.0f / (1.0f + __expf(-x)); }

// ---------------- q = x@Wq^T + bq ; kk = x@Wk^T + bk ----------------
__global__ void k_qk(const float* __restrict__ x, const float* __restrict__ Wq,
                     const float* __restrict__ bq, const float* __restrict__ Wk,
                     const float* __restrict__ bk, float* __restrict__ q,
                     float* __restrict__ kk, int L) {
    int r = blockIdx.x * blockDim.x + threadIdx.x;
    if (r >= L) return;
    float xv[E];
#pragma unroll
    for (int d = 0; d < E; ++d) xv[d] = x[r * E + d];
#pragma unroll
    for (int e = 0; e < E; ++e) {
        float a = bq[e], b = bk[e];
#pragma unroll
        for (int d = 0; d < E; ++d) {
            a += Wq[e * E + d] * xv[d];
            b += Wk[e * E + d] * xv[d];
        }
        q[r * E + e]  = a;
        kk[r * E + e] = b;
    }
}

// ---------------- h1 = x@W1^T + b1 via V_WMMA_F32_16X16X4_F32 ----------------
// One wave (32 threads) per 16-row tile. K=5 split as K=4 + zero-padded K=4.
// N=30 padded to two 16-wide tiles.
__global__ void k_h1_wmma(const float* __restrict__ x, const float* __restrict__ W1,
                          const float* __restrict__ b1v, float* __restrict__ h1, int L) {
    int row0 = blockIdx.x * 16;
    int lane = threadIdx.x & 31;
    int half = lane >> 4;   // 0: K 0..1 (or M 0..7 in C), 1: K 2..3 (or M 8..15)
    int lm   = lane & 15;

    int arow = row0 + lm;
    if (arow >= L) arow = L - 1;  // clamp (tail tile rows are masked on store)

    // A: 16x4 tile of x. VGPR v holds K = 2*half + v.
    v2f a0 = { x[arow * E + 2 * half], x[arow * E + 2 * half + 1] };
    // second K-step: global K = 4 + 2*half + v ; only K==4 is real.
    v2f a1 = { (half == 0) ? x[arow * E + 4] : 0.0f, 0.0f };

#pragma unroll
    for (int t = 0; t < 2; ++t) {
        int col = lm + 16 * t;  // output column (n)
        v2f b0 = { 0.0f, 0.0f }, b1 = { 0.0f, 0.0f };
        if (col < H) {
            // B[k][n] = W1[n][k]
            b0.x = W1[col * E + 2 * half];
            b0.y = W1[col * E + 2 * half + 1];
            if (half == 0) b1.x = W1[col * E + 4];
        }
        v8f c = {};
        c = __builtin_amdgcn_wmma_f32_16x16x4_f32(false, a0, false, b0, (short)0, c, false, false);
        c = __builtin_amdgcn_wmma_f32_16x16x4_f32(false, a1, false, b1, (short)0, c, false, false);

        float bias = (col < H) ? b1v[col] : 0.0f;
#pragma unroll
        for (int v = 0; v < 8; ++v) {
            int row = row0 + v + 8 * half;  // C/D layout: VGPR v -> M = v + 8*half, N = lm
            if (col < H && row < L) h1[row * H + col] = c[v] + bias;
        }
    }
}

// ---------------- invden(i,e) = 1 / sum_j exp(q[i,e]*kk[j,e]) ----------------
__global__ void k_denom(const float* __restrict__ q, const float* __restrict__ kk,
                        float* __restrict__ invden, int L) {
    extern __shared__ float skk[];  // L floats: column e of kk
    int e = blockIdx.y;
    for (int t = threadIdx.x; t < L; t += blockDim.x) skk[t] = kk[t * E + e];
    __syncthreads();
    int i = blockIdx.x * blockDim.x + threadIdx.x;
    if (i >= L) return;
    float qv = q[i * E + e];
    float s = 0.0f;
    for (int j = 0; j < L; ++j) s += __expf(qv * skk[j]);
    invden[i * E + e] = 1.0f / s;
}

// ---------------- mean_w[j] = 1/(L*E) sum_{i,e} exp(q*kk)*invden ----------------
__global__ void k_meanw(const float* __restrict__ q, const float* __restrict__ kk,
                        const float* __restrict__ invden, float* __restrict__ mw, int L) {
    const int CH = 64;
    __shared__ float sq[CH * E];
    __shared__ float si[CH * E];
    int j = blockIdx.x * blockDim.x + threadIdx.x;
    float kj[E];
#pragma unroll
    for (int e = 0; e < E; ++e) kj[e] = (j < L) ? kk[j * E + e] : 0.0f;
    float s = 0.0f;
    for (int i0 = 0; i0 < L; i0 += CH) {
        __syncthreads();
        for (int t = threadIdx.x; t < CH * E; t += blockDim.x) {
            sq[t] = q[i0 * E + t];
            si[t] = invden[i0 * E + t];
        }
        __builtin_prefetch(&q[(i0 + CH) * E], 0, 1);       // global_prefetch_b8
        __builtin_prefetch(&invden[(i0 + CH) * E], 0, 1);
        __syncthreads();
        for (int ii = 0; ii < CH; ++ii) {
#pragma unroll
            for (int e = 0; e < E; ++e)
                s += __expf(sq[ii * E + e] * kj[e]) * si[ii * E + e];
        }
    }
    if (j < L) mw[j] = s / (float)(L * E);
}

// ---------------- rank + select (top-k with lax.top_k tie-break) ----------------
__global__ void k_rank(const float* __restrict__ mw, int* __restrict__ flags, int L, int k) {
    __shared__ float sw[256];
    int j = blockIdx.x * blockDim.x + threadIdx.x;
    float v = (j < L) ? mw[j] : 0.0f;
    int cnt = 0;
    for (int b0 = 0; b0 < L; b0 += 256) {
        __syncthreads();
        int src = b0 + (int)threadIdx.x;
        sw[threadIdx.x] = (src < L) ? mw[src] : -__builtin_huge_valf();
        __syncthreads();
        for (int t = 0; t < 256; ++t) {
            int jp = b0 + t;
            float w = sw[t];
            cnt += ((w > v) || (w == v && jp < j)) ? 1 : 0;
        }
    }
    if (j < L) flags[j] = (cnt < k) ? 1 : 0;
}

// ---------------- ordered compaction of selected indices ----------------
__global__ void k_compact(const int* __restrict__ flags, int* __restrict__ idxb,
                          int* __restrict__ count, int L) {
    __shared__ int part[256];
    __shared__ int base[256];
    int tid = threadIdx.x;
    int seg = (L + 255) / 256;
    int s = 0;
    for (int t = 0; t < seg; ++t) {
        int j = tid * seg + t;
        if (j < L) s += flags[j];
    }
    part[tid] = s;
    __syncthreads();
    if (tid == 0) {
        int acc = 0;
        for (int t = 0; t < 256; ++t) { base[t] = acc; acc += part[t]; }
        count[0] = acc;
    }
    __syncthreads();
    int pos = base[tid];
    for (int t = 0; t < seg; ++t) {
        int j = tid * seg + t;
        if (j < L && flags[j]) idxb[pos++] = j;
    }
}

// ---------------- sequential LSTM over h1[idx]; writes h back into h1 ----------------
__global__ void k_lstm(const float* __restrict__ Wih, const float* __restrict__ Whh,
                       const float* __restrict__ bih, const float* __restrict__ bhh,
                       const int* __restrict__ idxb, float* __restrict__ h1, int nsteps) {
    __shared__ float swih[4 * H * H];
    __shared__ float swhh[4 * H * H];
    __shared__ float sbz[4 * H];
    __shared__ float sh[H], sc[H], sxt[H], sz[4 * H];
    int tid = threadIdx.x;  // 128 threads
    for (int t = tid; t < 4 * H * H; t += blockDim.x) { swih[t] = Wih[t]; swhh[t] = Whh[t]; }
    for (int t = tid; t < 4 * H; t += blockDim.x) sbz[t] = bih[t] + bhh[t];
    if (tid < H) { sh[tid] = 0.0f; sc[tid] = 0.0f; }
    __syncthreads();
    for (int s = 0; s < nsteps; ++s) {
        int row = idxb[s];
        if (tid < H) sxt[tid] = h1[row * H + tid];
        __syncthreads();
        if (tid < 4 * H) {
            float z = sbz[tid];
            const float* wi = &swih[tid * H];
            const float* wh = &swhh[tid * H];
#pragma unroll
            for (int d = 0; d < H; ++d) z += wi[d] * sxt[d] + wh[d] * sh[d];
            sz[tid] = z;
        }
        __syncthreads();
        if (tid < H) {
            float iv = sigf(sz[tid]);
            float fv = sigf(sz[H + tid]);
            float gv = tanhf(sz[2 * H + tid]);
            float ov = sigf(sz[3 * H + tid]);
            float cn = fv * sc[tid] + iv * gv;
            sc[tid] = cn;
            float hn = ov * tanhf(cn);
            sh[tid] = hn;
            h1[row * H + tid] = hn;
        }
        __syncthreads();
    }
}

// ---------------- out = combined @ W2^T + b2 ----------------
__global__ void k_out(const float* __restrict__ h1, const float* __restrict__ W2,
                      const float* __restrict__ b2, float* __restrict__ out, int L) {
    int r = blockIdx.x * blockDim.x + threadIdx.x;
    if (r >= L) return;
    float s = b2[0];
#pragma unroll
    for (int d = 0; d < H; ++d) s += h1[r * H + d] * W2[d];
    out[r] = s;
}

extern "C" void kernel_launch(void* const* d_in, const int* in_sizes, int n_in,
                              void* d_out, int out_size, void* d_ws, size_t ws_size,
                              hipStream_t stream) {
    const float* x   = (const float*)d_in[0];
    const float* Wq  = (const float*)d_in[1];
    const float* bq  = (const float*)d_in[2];
    const float* Wk  = (const float*)d_in[3];
    const float* bk  = (const float*)d_in[4];
    const float* W1  = (const float*)d_in[5];
    const float* b1  = (const float*)d_in[6];
    const float* Wih = (const float*)d_in[7];
    const float* Whh = (const float*)d_in[8];
    const float* bih = (const float*)d_in[9];
    const float* bhh = (const float*)d_in[10];
    const float* W2  = (const float*)d_in[11];
    const float* b2  = (const float*)d_in[12];
    float* out = (float*)d_out;

    int L = in_sizes[0] / E;
    int k = (int)(0.3 * (double)L);  // matches Python int(0.3*L) truncation

    char* w = (char*)d_ws;
    float* q   = (float*)w; w += (size_t)L * E * sizeof(float);
    float* kk  = (float*)w; w += (size_t)L * E * sizeof(float);
    float* inv = (float*)w; w += (size_t)L * E * sizeof(float);
    float* mw  = (float*)w; w += (size_t)L * sizeof(float);
    float* h1  = (float*)w; w += (size_t)L * H * sizeof(float);
    int* flags = (int*)w;   w += (size_t)L * sizeof(int);
    int* idxb  = (int*)w;   w += (size_t)L * sizeof(int);
    int* cnt   = (int*)w;   w += sizeof(int);
    (void)ws_size; (void)n_in; (void)out_size;

    int nb = (L + 255) / 256;
    k_qk<<<nb, 256, 0, stream>>>(x, Wq, bq, Wk, bk, q, kk, L);
    k_h1_wmma<<<(L + 15) / 16, 32, 0, stream>>>(x, W1, b1, h1, L);
    dim3 gden(nb, E);
    k_denom<<<gden, 256, (size_t)L * sizeof(float), stream>>>(q, kk, inv, L);
    k_meanw<<<nb, 256, 0, stream>>>(q, kk, inv, mw, L);
    k_rank<<<nb, 256, 0, stream>>>(mw, flags, L, k);
    k_compact<<<1, 256, 0, stream>>>(flags, idxb, cnt, L);
    k_lstm<<<1, 128, 0, stream>>>(Wih, Whh, bih, bhh, idxb, h1, k);
    k_out<<<nb, 256, 0, stream>>>(h1, W2, b2, out, L);
}